// KLLoss_26259430048220
// MI455X (gfx1250) — compile-verified
//
#include <hip/hip_runtime.h>

// KL(p||q) for diagonal Gaussians over (B=16, L=64, N=512, D=64) f32 tensors,
// reduced to a scalar: sum over all elems of
//   ln(qs) - ln(ps) + 0.5*(ps^2 + (pm-qm)^2)/qs^2 - 0.5
// divided by B*L = 1024.
//
// Memory-bound: 512 MiB streamed @ 23.3 TB/s => ~23 us floor. Strategy:
//   - float4 (global_load_b128) non-temporal streaming loads (no reuse, > L2)
//   - per-thread f32 accumulation (64 elems/thread: negligible rounding)
//   - wave reduction via V_WMMA_F32_16X16X4_F32 (A = lane sums, B = ones)
//   - block partials -> d_ws, second tiny kernel -> d_out (deterministic,
//     no atomics into non-repoisoned buffers)

typedef float v2f __attribute__((ext_vector_type(2)));
typedef float v4f __attribute__((ext_vector_type(4)));
typedef float v8f __attribute__((ext_vector_type(8)));

#define KL_NBLOCKS  2048
#define KL_NTHREADS 256
#define KL_NWAVES   (KL_NTHREADS / 32)   // 8 wave32 per block

// ln(qs) - ln(ps) + 0.5*(ps^2 + dmu^2) * (1/qs)^2 - 0.5
__device__ __forceinline__ float kl_elem(float pm, float ps, float qm, float qs) {
    const float LN2 = 0.69314718055994530942f;
    float r    = __builtin_amdgcn_rcpf(qs);                         // v_rcp_f32
    float dl   = (__builtin_amdgcn_logf(qs) -                      // v_log_f32 (log2)
                  __builtin_amdgcn_logf(ps)) * LN2;
    float dm   = pm - qm;
    float quad = __builtin_fmaf(ps, ps, dm * dm);                  // ps^2 + dmu^2
    return __builtin_fmaf(quad, 0.5f * r * r, dl - 0.5f);
}

// Exact block-sum. Wave32 lane partials are collapsed with one
// v_wmma_f32_16x16x4_f32:
//   A vgpr0 = s (lane sum), A vgpr1 = 0  => A[m][0]=s[m], A[m][2]=s[m+16]
//   B = ones                             => D[m][n] = s[m] + s[m+16]
// Summing the 8 D vgprs: lane 0 holds sum of lanes {0..7,16..23},
// lane 16 holds sum of lanes {8..15,24..31}. Two LDS stores per wave,
// thread 0 adds 2*NWAVES values. Result valid in thread 0 only.
__device__ __forceinline__ float block_reduce_wmma(float s) {
    __shared__ float lds[2 * KL_NWAVES];
    const int lane = threadIdx.x & 31;
    const int wave = threadIdx.x >> 5;

    v2f a; a[0] = s;    a[1] = 0.0f;
    v2f b; b[0] = 1.0f; b[1] = 1.0f;
    v8f c = {};
    // 8 args: (neg_a, A, neg_b, B, c_mod, C, reuse_a, reuse_b)
    c = __builtin_amdgcn_wmma_f32_16x16x4_f32(
            /*neg_a=*/false, a, /*neg_b=*/false, b,
            /*c_mod=*/(short)0, c, /*reuse_a=*/false, /*reuse_b=*/false);

    float v = ((c[0] + c[1]) + (c[2] + c[3])) + ((c[4] + c[5]) + (c[6] + c[7]));
    if (lane == 0)  lds[2 * wave]     = v;   // rows 0..7  -> lanes 0..7,16..23
    if (lane == 16) lds[2 * wave + 1] = v;   // rows 8..15 -> lanes 8..15,24..31
    __syncthreads();

    float tot = 0.0f;
    if (threadIdx.x == 0) {
        #pragma unroll
        for (int i = 0; i < 2 * KL_NWAVES; ++i) tot += lds[i];
    }
    return tot;
}

__global__ void __launch_bounds__(KL_NTHREADS)
kl_partial_kernel(const float* __restrict__ pm, const float* __restrict__ ps,
                  const float* __restrict__ qm, const float* __restrict__ qs,
                  float* __restrict__ partial, long n4) {
    const v4f* __restrict__ pm4 = (const v4f*)pm;
    const v4f* __restrict__ ps4 = (const v4f*)ps;
    const v4f* __restrict__ qm4 = (const v4f*)qm;
    const v4f* __restrict__ qs4 = (const v4f*)qs;

    const long tid    = (long)blockIdx.x * blockDim.x + threadIdx.x;
    const long stride = (long)gridDim.x * blockDim.x;

    float acc = 0.0f;
    #pragma unroll 4
    for (long i = tid; i < n4; i += stride) {
        // Streaming (non-temporal) 128-bit loads: data has zero reuse and
        // exceeds L2; keep it out of the cache hierarchy.
        v4f a = __builtin_nontemporal_load(pm4 + i);
        v4f b = __builtin_nontemporal_load(ps4 + i);
        v4f c = __builtin_nontemporal_load(qm4 + i);
        v4f d = __builtin_nontemporal_load(qs4 + i);
        acc += kl_elem(a[0], b[0], c[0], d[0]);
        acc += kl_elem(a[1], b[1], c[1], d[1]);
        acc += kl_elem(a[2], b[2], c[2], d[2]);
        acc += kl_elem(a[3], b[3], c[3], d[3]);
    }

    float tot = block_reduce_wmma(acc);
    if (threadIdx.x == 0) partial[blockIdx.x] = tot;   // overwritten every call
}

__global__ void __launch_bounds__(KL_NTHREADS)
kl_final_kernel(const float* __restrict__ partial, float* __restrict__ out,
                int n, float scale) {
    float acc = 0.0f;
    for (int i = threadIdx.x; i < n; i += blockDim.x) acc += partial[i];
    float tot = block_reduce_wmma(acc);
    if (threadIdx.x == 0) out[0] = tot * scale;
}

extern "C" void kernel_launch(void* const* d_in, const int* in_sizes, int n_in,
                              void* d_out, int out_size, void* d_ws, size_t ws_size,
                              hipStream_t stream) {
    // setup_inputs() dict order: prior_mu, prior_sigma, post_mu, post_sigma
    const float* prior_mu    = (const float*)d_in[0];
    const float* prior_sigma = (const float*)d_in[1];
    const float* post_mu     = (const float*)d_in[2];
    const float* post_sigma  = (const float*)d_in[3];
    float* out     = (float*)d_out;
    float* partial = (float*)d_ws;        // KL_NBLOCKS floats (8 KiB) scratch

    const long n  = (long)in_sizes[0];    // 16*64*512*64 = 33,554,432
    const long n4 = n >> 2;               // float4 quads

    kl_partial_kernel<<<KL_NBLOCKS, KL_NTHREADS, 0, stream>>>(
        prior_mu, prior_sigma, post_mu, post_sigma, partial, n4);

    // mean over (B, L) = divide by 16*64 = 1024
    kl_final_kernel<<<1, KL_NTHREADS, 0, stream>>>(
        partial, out, KL_NBLOCKS, 1.0f / 1024.0f);
}